// PanopticHead_71270687310520
// MI455X (gfx1250) — compile-verified
//
#include <hip/hip_runtime.h>
#include <cstdint>

// Problem constants (from reference)
#define N_INST 64
#define MM     28
#define THING  80
#define STUFF  53
#define HH     512
#define WW     800
#define HW     (HH * WW)          // 409600 pixels per channel
#define TILE   4096               // pixels per block
#define BLOCKS_PER_CH (HW / TILE) // 100
#define NTHREADS 256              // 8 wave32s

typedef float v4f __attribute__((ext_vector_type(4)));   // native vector -> NT builtins OK

__global__ __launch_bounds__(NTHREADS)
void panoptic_head_kernel(const float* __restrict__ mask_logits,   // [64,80,28,28]
                          const float* __restrict__ sem,           // [133,512,800]
                          const float* __restrict__ bboxes,        // [64,4]
                          const int*   __restrict__ cls_idx,       // [64]
                          float*       __restrict__ out)           // [117,512,800]
{
    const int bid  = blockIdx.x;
    const int c    = bid / BLOCKS_PER_CH;        // output channel 0..116
    const int tile = bid - c * BLOCKS_PER_CH;
    const int base = tile * TILE;                // pixel offset within channel
    const int tid  = threadIdx.x;

    if (c < STUFF) {
        // ---- stuff channels: streaming b128 copy, NT both ways ----
        // (read-once source, write-once dest: keep both out of L2 so the
        //  reusable mask/sem working set stays resident)
        const v4f* __restrict__ src = (const v4f*)(sem + (size_t)c * HW + base);
        v4f*       __restrict__ dst = (v4f*)(out + (size_t)c * HW + base);
        #pragma unroll
        for (int k = 0; k < TILE / 4 / NTHREADS; ++k) {
            v4f v = __builtin_nontemporal_load(src + tid + k * NTHREADS);
            __builtin_nontemporal_store(v, dst + tid + k * NTHREADS);
        }
        return;
    }

    // ---- thing channels ----
    const int n   = c - STUFF;
    const int cls = cls_idx[n];

    const float bx1 = bboxes[n * 4 + 0];
    const float by1 = bboxes[n * 4 + 1];
    const float bx2 = bboxes[n * 4 + 2];
    const float by2 = bboxes[n * 4 + 3];

    const int x0 = (int)floorf(bx1);
    const int y0 = (int)floorf(by1);
    const int x2 = (int)floorf(bx2);
    const int y2 = (int)floorf(by2);
    const float inv_bw = (float)MM / (float)(x2 - x0 + 1);
    const float inv_bh = (float)MM / (float)(y2 - y0 + 1);

    // mask-paste box (clamped to image)
    const int mx_lo = max(x0, 0), mx_hi = min(x2 + 1, WW);
    const int my_lo = max(y0, 0), my_hi = min(y2 + 1, HH);
    // sem-crop box (floor / round-half-even per jnp.floor / jnp.round)
    const int cx1 = (int)floorf(bx1);
    const int cy1 = (int)floorf(by1);
    const int cx2 = (int)rintf(bx2) + 1;
    const int cy2 = (int)rintf(by2) + 1;

    // ---- stage the 28x28 instance mask into LDS via CDNA5 async copy ----
    __shared__ __align__(16) float smem[MM * MM];   // 784 f32 = 3136 B
    {
        const float* msrc = mask_logits + ((size_t)n * THING + (size_t)cls) * (MM * MM);
        const uint32_t lds0 = (uint32_t)(uintptr_t)(&smem[0]);   // low 32b = LDS byte offset
        constexpr int NVEC = (MM * MM) / 4;                      // 196 x b128
        if (tid < NVEC) {
            uint32_t      la = lds0 + (uint32_t)tid * 16u;
            const float*  ga = msrc + tid * 4;
            asm volatile("global_load_async_to_lds_b128 %0, %1, off"
                         :: "v"(la), "v"(ga)
                         : "memory");
        }
        asm volatile("s_wait_asynccnt 0" ::: "memory");
        __syncthreads();
    }

    const float* __restrict__ tsem = sem + (size_t)(STUFF + cls) * HW;
    float*       __restrict__ dst  = out + (size_t)c * HW + base;

    #pragma unroll
    for (int k = 0; k < TILE / 4 / NTHREADS; ++k) {
        const int vi = tid + k * NTHREADS;    // float4 index within tile
        const int p  = base + vi * 4;         // pixel index within channel
        const int h  = p / WW;
        const int w  = p - h * WW;            // 800 % 4 == 0 -> all 4 pixels share row h

        // y-direction terms are uniform across the 4 pixels
        const bool in_y  = (h >= my_lo) & (h < my_hi);
        const bool cin_y = (h >= cy1) & (h < cy2);
        float sy = ((float)h - (float)y0 + 0.5f) * inv_bh - 0.5f;
        sy = fminf(fmaxf(sy, 0.0f), (float)(MM - 1));
        const int   iy0 = (int)sy;                 // sy >= 0 -> trunc == floor
        const int   iy1 = min(iy0 + 1, MM - 1);
        const float ly  = sy - (float)iy0;
        const float oly = 1.0f - ly;

        v4f v;
        #pragma unroll
        for (int j = 0; j < 4; ++j) {
            const int x = w + j;
            float val = 0.0f;
            if (in_y & (x >= mx_lo) & (x < mx_hi)) {
                float sx = ((float)x - (float)x0 + 0.5f) * inv_bw - 0.5f;
                sx = fminf(fmaxf(sx, 0.0f), (float)(MM - 1));
                const int   ix0 = (int)sx;
                const int   ix1 = min(ix0 + 1, MM - 1);
                const float lx  = sx - (float)ix0;
                // match reference order: lerp in y first, then x
                const float f0 = smem[iy0 * MM + ix0] * oly + smem[iy1 * MM + ix0] * ly;
                const float f1 = smem[iy0 * MM + ix1] * oly + smem[iy1 * MM + ix1] * ly;
                val = f0 * (1.0f - lx) + f1 * lx;
            }
            if (cin_y & (x >= cx1) & (x < cx2))
                val += tsem[h * WW + x];
            v[j] = val;
        }
        __builtin_nontemporal_store(v, ((v4f*)dst) + vi);
    }
}

extern "C" void kernel_launch(void* const* d_in, const int* in_sizes, int n_in,
                              void* d_out, int out_size, void* d_ws, size_t ws_size,
                              hipStream_t stream) {
    const float* mask_logits = (const float*)d_in[0];
    const float* sem         = (const float*)d_in[1];
    const float* bboxes      = (const float*)d_in[2];
    const int*   cls_idx     = (const int*)d_in[3];
    float*       out         = (float*)d_out;

    const int grid = (STUFF + N_INST) * BLOCKS_PER_CH;   // 117 * 100 = 11700 blocks
    panoptic_head_kernel<<<grid, NTHREADS, 0, stream>>>(mask_logits, sem, bboxes,
                                                        cls_idx, out);
}